// PINO1d_53781580480553
// MI455X (gfx1250) — compile-verified
//
#include <hip/hip_runtime.h>
#include <math.h>

// ---------------- types for WMMA f32 16x16x4 ----------------
typedef float v2f __attribute__((ext_vector_type(2)));
typedef float v8f __attribute__((ext_vector_type(8)));

__device__ __forceinline__ v8f wmma4(v2f a, v2f b, v8f c) {
    // (neg_a, A, neg_b, B, c_mod, C, reuse_a, reuse_b)
    return __builtin_amdgcn_wmma_f32_16x16x4_f32(false, a, false, b, (short)0, c, false, false);
}

__device__ __forceinline__ float gelu_f(float v) {
    return 0.5f * v * (1.0f + erff(v * 0.70710678118654752440f));
}

#define BATCH 64
#define WID   64
#define NPT   8192
#define NMODE 16
#define JDIM  32   // re/im interleaved modes

// CDNA5 async global->LDS path (gfx1250): guarded so missing builtins fall
// back to the synchronous staging path instead of breaking the build.
#if defined(__gfx1250__) && __has_builtin(__builtin_amdgcn_global_load_async_to_lds_b128) && __has_builtin(__builtin_amdgcn_s_wait_asynccnt)
#define FNO_ASYNC 1
// builtin takes (addrspace(1) int4* src, addrspace(3) int4* dst, imm off, imm cpol)
typedef int b128v __attribute__((vector_size(4 * sizeof(int))));
typedef __attribute__((address_space(1))) b128v* g128p;
typedef __attribute__((address_space(3))) b128v* l128p;
#endif

// ---------------------------------------------------------------------------
// K1: build forward basis F[n][32] and inverse basis Ginv[n][32]
//   F[n][2m]   =  cos(2*pi*m*n/N)      F[n][2m+1]   = -sin(2*pi*m*n/N)
//   Ginv[n][2m]=  s_m*cos(...)         Ginv[n][2m+1]= -s_m*sin(...)  (0 for m==0)
//   s_0 = 1/N, s_m = 2/N  (irfft of 16-mode spectrum)
// ---------------------------------------------------------------------------
__global__ void basis_kernel(float* __restrict__ F, float* __restrict__ Ginv) {
    int n = blockIdx.x * blockDim.x + threadIdx.x;
    if (n >= NPT) return;
    const float w0 = 6.28318530717958647692f / (float)NPT;
    for (int m = 0; m < NMODE; ++m) {
        int r = (m * n) & (NPT - 1);          // exact integer angle reduction
        float ang = w0 * (float)r;
        float s, c;
        __sincosf(ang, &s, &c);
        F[n * JDIM + 2 * m]     = c;
        F[n * JDIM + 2 * m + 1] = -s;
        float sm = (m == 0) ? (1.0f / (float)NPT) : (2.0f / (float)NPT);
        Ginv[n * JDIM + 2 * m]     = sm * c;
        Ginv[n * JDIM + 2 * m + 1] = (m == 0) ? 0.0f : -sm * s;
    }
}

// ---------------------------------------------------------------------------
// K2: lift  h[b][w][n] = sum_i x[b][n][i]*lift_w[i][w] + lift_b[w]
// ---------------------------------------------------------------------------
__global__ void lift_kernel(const float* __restrict__ x,
                            const float* __restrict__ lw,
                            const float* __restrict__ lb,
                            float* __restrict__ h) {
    long idx = (long)blockIdx.x * blockDim.x + threadIdx.x;  // over B*W*N
    int n = (int)(idx & (NPT - 1));
    int w = (int)((idx >> 13) & (WID - 1));
    int b = (int)(idx >> 19);
    const float* xp = x + ((long)b * NPT + n) * 3;
    float v = xp[0] * lw[0 * WID + w] + xp[1] * lw[1 * WID + w] + xp[2] * lw[2 * WID + w] + lb[w];
    h[idx] = v;
}

// ---------------------------------------------------------------------------
// K3: forward transform  Xf[r][j] = sum_n H[r][n] * F[n][j]
//   r = b*64+c in [0,4096), j in [0,32).  M=4096, N=32, K=8192.
//   Block: 256 threads = 8 waves.  wave w: jtile = w&1, kslice = w>>1 (2048 K).
//   Wave-pairs (same kslice) share a double-buffered LDS-staged 16x32 A chunk
//   filled with async global->LDS b128 copies (ASYNCcnt) overlapping the WMMAs.
//   Row pad = 36 floats: 36*r mod 64 distinct for r=0..15 (conflict-free A
//   fragment reads) and 144B row stride keeps every 16B segment aligned.
// ---------------------------------------------------------------------------
__global__ void fwd_kernel(const float* __restrict__ H,
                           const float* __restrict__ F,
                           float* __restrict__ Xf) {
    __shared__ float stage[2][4][16][36];  // [buf][kslice pair][Mrow][32 k + pad]
    __shared__ float part[8][256];         // per-wave partial C tiles

    const int tid  = threadIdx.x;
    const int wv   = tid >> 5;           // wave 0..7
    const int lane = tid & 31;
    const int l15  = lane & 15;
    const int half = lane >> 4;
    const int jt   = wv & 1;             // j tile
    const int ks   = wv >> 1;            // k slice 0..3
    const int r0   = blockIdx.x * 16;    // row tile base
    const int pr   = tid >> 6;           // pair id 0..3 (64 threads each)
    const int pt   = tid & 63;           // thread within pair

    const int ks0 = ks * 2048;
    v8f acc = {0.f, 0.f, 0.f, 0.f, 0.f, 0.f, 0.f, 0.f};

#ifdef FNO_ASYNC
    // each thread fills 2 x 16B segments of its pair's 16x32 chunk
    const int rowA0 = pt >> 3,        segA0 = pt & 7;
    const int rowA1 = (pt + 64) >> 3, segA1 = (pt + 64) & 7;
    #define FWD_ISSUE(kc_, buf_)                                                     \
        do {                                                                         \
            const float* g0 = H + (long)(r0 + rowA0) * NPT + pr * 2048 + (kc_) + segA0 * 4; \
            const float* g1 = H + (long)(r0 + rowA1) * NPT + pr * 2048 + (kc_) + segA1 * 4; \
            __builtin_amdgcn_global_load_async_to_lds_b128(                          \
                (g128p)g0, (l128p)&stage[buf_][pr][rowA0][segA0 * 4], 0, 0);         \
            __builtin_amdgcn_global_load_async_to_lds_b128(                          \
                (g128p)g1, (l128p)&stage[buf_][pr][rowA1][segA1 * 4], 0, 0);         \
        } while (0)

    int buf = 0;
    FWD_ISSUE(0, 0);
    for (int kc = 0; kc < 2048; kc += 32) {
        __builtin_amdgcn_s_wait_asynccnt(0);
        __syncthreads();
        if (kc + 32 < 2048) FWD_ISSUE(kc + 32, buf ^ 1);
        const float* Fp = F + (long)(ks0 + kc) * JDIM + jt * 16 + l15;
        #pragma unroll
        for (int kk = 0; kk < 8; ++kk) {
            int kb = kk * 4 + 2 * half;
            v2f a, b;
            a.x = stage[buf][ks][l15][kb];
            a.y = stage[buf][ks][l15][kb + 1];
            b.x = Fp[(long)kb * JDIM];
            b.y = Fp[(long)(kb + 1) * JDIM];
            acc = wmma4(a, b, acc);
        }
        buf ^= 1;
    }
    #undef FWD_ISSUE
#else
    for (int kc = 0; kc < 2048; kc += 32) {
        __syncthreads();
        for (int q = 0; q < 8; ++q) {
            int idx = pt + 64 * q;              // 0..511
            int row = idx >> 5;
            int col = idx & 31;
            stage[0][pr][row][col] = H[(long)(r0 + row) * NPT + (pr * 2048 + kc + col)];
        }
        __syncthreads();
        const float* Fp = F + (long)(ks0 + kc) * JDIM + jt * 16 + l15;
        #pragma unroll
        for (int kk = 0; kk < 8; ++kk) {
            int kb = kk * 4 + 2 * half;
            v2f a, b;
            a.x = stage[0][ks][l15][kb];
            a.y = stage[0][ks][l15][kb + 1];
            b.x = Fp[(long)kb * JDIM];
            b.y = Fp[(long)(kb + 1) * JDIM];
            acc = wmma4(a, b, acc);
        }
    }
#endif

    // dump partial tiles: element (M = v + 8*half, n = l15)
    #pragma unroll
    for (int v = 0; v < 8; ++v)
        part[wv][(v + 8 * half) * 16 + l15] = acc[v];
    __syncthreads();

    if (wv < 2) {  // wave wv reduces jtile==wv over kslices
        for (int e = lane; e < 256; e += 32) {
            float s = part[wv][e] + part[wv + 2][e] + part[wv + 4][e] + part[wv + 6][e];
            int M = e >> 4, nn = e & 15;
            Xf[(long)(r0 + M) * JDIM + wv * 16 + nn] = s;
        }
    }
}

// ---------------------------------------------------------------------------
// K4: complex mode mix  Yf[b][o][m] = sum_c Xf[b][c][m] * w[c][o][m]
// ---------------------------------------------------------------------------
__global__ void mix_kernel(const float* __restrict__ Xf,
                           const float* __restrict__ sw,
                           float* __restrict__ Yf) {
    int m = blockIdx.x & (NMODE - 1);
    int b = blockIdx.x >> 4;
    int o = threadIdx.x;
    float yr = 0.f, yi = 0.f;
    for (int c = 0; c < WID; ++c) {
        float xr = Xf[((long)b * WID + c) * JDIM + 2 * m];
        float xi = Xf[((long)b * WID + c) * JDIM + 2 * m + 1];
        const float* wp = sw + (((long)c * WID + o) * NMODE + m) * 2;
        float wr = wp[0], wi = wp[1];
        yr += xr * wr - xi * wi;
        yi += xr * wi + xi * wr;
    }
    Yf[((long)b * WID + o) * JDIM + 2 * m]     = yr;
    Yf[((long)b * WID + o) * JDIM + 2 * m + 1] = yi;
}

// ---------------------------------------------------------------------------
// K5: fused block  out[b][o][n] = gelu( [W | Yf_b] x [in_b ; Ginv^T] + bias )
//   K = 96 (64 conv + 32 inverse-DFT).  Block = 128 thr = 4 waves (one o-tile
//   each); B'[96][16] tile staged in LDS, shared by all 4 waves.
// ---------------------------------------------------------------------------
__global__ void block_kernel(const float* __restrict__ in,
                             const float* __restrict__ Yf,
                             const float* __restrict__ Ginv,
                             const float* __restrict__ Wm,
                             const float* __restrict__ bias,
                             float* __restrict__ out) {
    __shared__ float bt[96 * 16];

    const int blk = blockIdx.x;
    const int b   = blk >> 9;            // / 512
    const int n0  = (blk & 511) * 16;
    const int tid = threadIdx.x;
    const int wv  = tid >> 5;
    const int lane = tid & 31;
    const int l15 = lane & 15;
    const int half = lane >> 4;
    const int o0  = wv * 16;

    // stage B' tile: rows 0..63 = in[b][k][n0+j], rows 64..95 = Ginv[n0+j][k-64]
    for (int idx = tid; idx < 96 * 16; idx += 128) {
        int k = idx >> 4, j = idx & 15;
        bt[idx] = (k < 64) ? in[((long)b * WID + k) * NPT + n0 + j]
                           : Ginv[(long)(n0 + j) * JDIM + (k - 64)];
    }
    __syncthreads();

    const int arow = o0 + l15;
    v8f acc = {0.f, 0.f, 0.f, 0.f, 0.f, 0.f, 0.f, 0.f};
    #pragma unroll
    for (int kk = 0; kk < 24; ++kk) {
        int kb = kk * 4 + 2 * half;
        v2f a, bfr;
        if (kk < 16) {  // conv part: A = W[o][i]
            a.x = Wm[(long)arow * WID + kb];
            a.y = Wm[(long)arow * WID + kb + 1];
        } else {        // spectral part: A = Yf[b][o][j]
            a.x = Yf[((long)b * WID + arow) * JDIM + (kb - 64)];
            a.y = Yf[((long)b * WID + arow) * JDIM + (kb - 63)];
        }
        bfr.x = bt[kb * 16 + l15];
        bfr.y = bt[(kb + 1) * 16 + l15];
        acc = wmma4(a, bfr, acc);
    }

    #pragma unroll
    for (int v = 0; v < 8; ++v) {
        int o = o0 + v + 8 * half;
        float r = gelu_f(acc[v] + bias[o]);
        out[((long)b * WID + o) * NPT + n0 + l15] = r;
    }
}

// ---------------------------------------------------------------------------
// K7: fused projection
//   q[k][n] = gelu( sum_w p1_w[w][k]*h[b][w][n] + p1_b[k] ),  k in [0,128)
//   out[b][n] = sum_k q[k][n]*p2_w[k] + p2_b
//   Block = 256 thr = 8 waves (one 16-row k-tile each); LDS atomic reduction.
// ---------------------------------------------------------------------------
__global__ void proj_kernel(const float* __restrict__ h,
                            const float* __restrict__ p1w,
                            const float* __restrict__ p1b,
                            const float* __restrict__ p2w,
                            const float* __restrict__ p2b,
                            float* __restrict__ out) {
    __shared__ float ht[64 * 16];
    __shared__ float red[16];

    const int blk = blockIdx.x;
    const int b   = blk >> 9;
    const int n0  = (blk & 511) * 16;
    const int tid = threadIdx.x;
    const int wv  = tid >> 5;
    const int lane = tid & 31;
    const int l15 = lane & 15;
    const int half = lane >> 4;
    const int k0  = wv * 16;

    for (int idx = tid; idx < 64 * 16; idx += 256) {
        int k = idx >> 4, j = idx & 15;
        ht[idx] = h[((long)b * WID + k) * NPT + n0 + j];
    }
    if (tid < 16) red[tid] = 0.f;
    __syncthreads();

    const int arow = k0 + l15;
    v8f acc = {0.f, 0.f, 0.f, 0.f, 0.f, 0.f, 0.f, 0.f};
    #pragma unroll
    for (int kk = 0; kk < 16; ++kk) {
        int wb = kk * 4 + 2 * half;
        v2f a, bfr;
        a.x = p1w[(long)wb * 128 + arow];        // p1_w[w][k] transposed access
        a.y = p1w[(long)(wb + 1) * 128 + arow];
        bfr.x = ht[wb * 16 + l15];
        bfr.y = ht[(wb + 1) * 16 + l15];
        acc = wmma4(a, bfr, acc);
    }

    float partial = 0.f;
    #pragma unroll
    for (int v = 0; v < 8; ++v) {
        int k = k0 + v + 8 * half;
        float q = gelu_f(acc[v] + p1b[k]);
        partial += q * p2w[k];
    }
    atomicAdd(&red[l15], partial);
    __syncthreads();

    if (tid < 16) out[(long)b * NPT + n0 + tid] = red[tid] + p2b[0];
}

// ---------------------------------------------------------------------------
extern "C" void kernel_launch(void* const* d_in, const int* in_sizes, int n_in,
                              void* d_out, int out_size, void* d_ws, size_t ws_size,
                              hipStream_t stream) {
    const float* x      = (const float*)d_in[0];
    const float* lift_w = (const float*)d_in[1];
    const float* lift_b = (const float*)d_in[2];
    const float* sw0    = (const float*)d_in[3];
    const float* sw1    = (const float*)d_in[4];
    const float* w0_w   = (const float*)d_in[5];
    const float* w0_b   = (const float*)d_in[6];
    const float* w1_w   = (const float*)d_in[7];
    const float* w1_b   = (const float*)d_in[8];
    const float* p1_w   = (const float*)d_in[9];
    const float* p1_b   = (const float*)d_in[10];
    const float* p2_w   = (const float*)d_in[11];
    const float* p2_b   = (const float*)d_in[12];
    float* out = (float*)d_out;

    float* ws = (float*)d_ws;
    const size_t ACT = (size_t)BATCH * WID * NPT;        // 33,554,432 floats
    float* bufA = ws;
    float* bufB = ws + ACT;
    float* F    = ws + 2 * ACT;
    float* Ginv = F + (size_t)NPT * JDIM;
    float* Xf   = Ginv + (size_t)NPT * JDIM;
    float* Yf   = Xf + (size_t)BATCH * WID * JDIM;

    basis_kernel<<<NPT / 256, 256, 0, stream>>>(F, Ginv);
    lift_kernel<<<(int)(ACT / 256), 256, 0, stream>>>(x, lift_w, lift_b, bufA);

    // block 0
    fwd_kernel<<<(BATCH * WID) / 16, 256, 0, stream>>>(bufA, F, Xf);
    mix_kernel<<<BATCH * NMODE, WID, 0, stream>>>(Xf, sw0, Yf);
    block_kernel<<<BATCH * (NPT / 16), 128, 0, stream>>>(bufA, Yf, Ginv, w0_w, w0_b, bufB);

    // block 1
    fwd_kernel<<<(BATCH * WID) / 16, 256, 0, stream>>>(bufB, F, Xf);
    mix_kernel<<<BATCH * NMODE, WID, 0, stream>>>(Xf, sw1, Yf);
    block_kernel<<<BATCH * (NPT / 16), 128, 0, stream>>>(bufB, Yf, Ginv, w1_w, w1_b, bufA);

    // projection
    proj_kernel<<<BATCH * (NPT / 16), 256, 0, stream>>>(bufA, p1_w, p1_b, p2_w, p2_b, out);
}